// RL2_ActorGNN_40767829574204
// MI455X (gfx1250) — compile-verified
//
#include <hip/hip_runtime.h>
#include <hip/hip_bf16.h>

// ---------------------------------------------------------------------------
// RL2 ActorGNN pipeline for MI455X (gfx1250, wave32).
// Dense projections -> V_WMMA_F32_16X16X4_F32 (full fp32 WMMA), with
//   compile-time K/F so all loads fold to immediate offsets, and 4 row tiles
//   per wave so each B fragment feeds 4 WMMAs.
// Edge softmax/aggregation -> L2-resident float atomics (25MB << 192MB L2).
// GRU -> VALU + LDS-staged weights (recurrence, wrong shape for WMMA).
// ---------------------------------------------------------------------------

typedef __attribute__((ext_vector_type(2))) float v2f;
typedef __attribute__((ext_vector_type(8))) float v8f;

#define GRU_T 50

static __device__ inline void edge_sd(const int* __restrict__ ei, int E, int e,
                                      int& s, int& d) {
  if (e < E) { s = ei[e]; d = ei[E + e]; }
  else       { s = e - E; d = e - E; }          // self loops appended
}

static __device__ inline void atomicMaxFloat(float* addr, float value) {
  // signed/unsigned trick: works for all non-NaN floats, init = -inf
  if (value >= 0.0f) atomicMax((int*)addr, __float_as_int(value));
  else               atomicMin((unsigned int*)addr, __float_as_uint(value));
}

// ---------------------------------------------------------------------------
// WMMA fp32 GEMM: Y[M,FF] = act(A[M,KK] @ W[KK,FF] + b[FF])
// ACT: 0 = none, 1 = relu.  KK % 4 == 0, FF % 16 == 0 (compile-time).
// Each wave computes a 64x16 strip: 4 row sub-tiles sharing one B fragment
// per K-step (4x B reuse). 4 waves (256 rows) per block.
// A layout (16x4 f32): lanes 0-15 rows, VGPR0/1 = K {0,1} (lanes<16) / {2,3}.
// B layout (4x16 f32): VGPR0/1 = K rows {0,1}/{2,3} striped, N = lane&15.
// C/D layout: VGPR v -> row v (lanes 0-15) / row v+8 (lanes 16-31), N=lane&15.
// ---------------------------------------------------------------------------
template <int ACT, int KK, int FF>
__global__ void wmma_gemm_kernel(const float* __restrict__ A,
                                 const float* __restrict__ W,
                                 const float* __restrict__ bias,
                                 float* __restrict__ Y,
                                 int Mrows) {
  const int lane = threadIdx.x & 31;
  const int wave = threadIdx.x >> 5;
  const int rowBase = (blockIdx.x * 4 + wave) * 64;
  if (rowBase >= Mrows) return;                 // wave-uniform exit
  const int col0 = blockIdx.y * 16;
  const int rl   = lane & 15;
  const int kh   = (lane >> 4) << 1;            // 0 or 2 (K half per lane group)
  const int c    = col0 + rl;

  // Per-sub-tile row pointers (clamped for tail; EXEC stays full for WMMA).
  const float* __restrict__ Ar[4];
#pragma unroll
  for (int s = 0; s < 4; ++s) {
    int r = rowBase + s * 16 + rl;
    if (r > Mrows - 1) r = Mrows - 1;
    Ar[s] = A + (size_t)r * KK + kh;
  }
  const float* __restrict__ Wp = W + (size_t)kh * FF + c;

  v8f acc[4];
#pragma unroll
  for (int s = 0; s < 4; ++s) acc[s] = (v8f){0.f,0.f,0.f,0.f,0.f,0.f,0.f,0.f};

#pragma unroll
  for (int k = 0; k < KK; k += 4) {
    v2f b;
    b[0] = Wp[k * FF];                 // K row k+kh
    b[1] = Wp[k * FF + FF];            // K row k+kh+1
#pragma unroll
    for (int s = 0; s < 4; ++s) {
      v2f a;
      a[0] = Ar[s][k];
      a[1] = Ar[s][k + 1];
      acc[s] = __builtin_amdgcn_wmma_f32_16x16x4_f32(
          /*neg_a=*/false, a, /*neg_b=*/false, b,
          /*c_mod=*/(short)0, acc[s], /*reuse_a=*/false, /*reuse_b=*/false);
    }
  }

  const float bc = bias[c];
#pragma unroll
  for (int s = 0; s < 4; ++s) {
    const int rb = rowBase + s * 16 + ((lane >> 4) << 3);
#pragma unroll
    for (int v = 0; v < 8; ++v) {
      const int rr = rb + v;
      if (rr < Mrows) {
        float val = acc[s][v] + bc;
        if (ACT == 1) val = fmaxf(val, 0.0f);
        Y[(size_t)rr * FF + c] = val;
      }
    }
  }
}

// ---------------------------------------------------------------------------
// GRU: one thread per sequence, hidden state in LDS (avoids dynamic-index
// register spills), weights staged in LDS. blockDim must be 128.
// ---------------------------------------------------------------------------
__global__ void gru_kernel(const float* __restrict__ seq,
                           const float* __restrict__ Wi,
                           const float* __restrict__ Wh,
                           const float* __restrict__ bi,
                           const float* __restrict__ bh,
                           float* __restrict__ Ht, int Mseq) {
  __shared__ float sWi[192];
  __shared__ float sWh[3072];
  __shared__ float sbi[96];
  __shared__ float sbh[96];
  __shared__ float hbuf[2][128 * 32];

  for (int i = threadIdx.x; i < 3072; i += blockDim.x) sWh[i] = Wh[i];
  for (int i = threadIdx.x; i < 192; i += blockDim.x) sWi[i] = Wi[i];
  for (int i = threadIdx.x; i < 96; i += blockDim.x) { sbi[i] = bi[i]; sbh[i] = bh[i]; }
  const int tid = threadIdx.x;
  for (int k = 0; k < 32; ++k) hbuf[0][tid * 32 + k] = 0.0f;
  __syncthreads();

  const int i = blockIdx.x * blockDim.x + tid;
  const bool active = (i < Mseq);
  int cur = 0;
  for (int t = 0; t < GRU_T; ++t) {
    float x0 = 0.f, x1 = 0.f;
    if (active) {
      x0 = seq[((size_t)i * GRU_T + t) * 2 + 0];
      x1 = seq[((size_t)i * GRU_T + t) * 2 + 1];
    }
    float* hc = &hbuf[cur][tid * 32];
    float* hn = &hbuf[1 - cur][tid * 32];
    for (int cch = 0; cch < 32; ++cch) {
      float hr = sbh[cch], hz = sbh[32 + cch], hnn = sbh[64 + cch];
      for (int k = 0; k < 32; ++k) {
        const float hk = hc[k];
        hr  = fmaf(hk, sWh[k * 96 + cch],      hr);
        hz  = fmaf(hk, sWh[k * 96 + 32 + cch], hz);
        hnn = fmaf(hk, sWh[k * 96 + 64 + cch], hnn);
      }
      const float ir  = sbi[cch]      + x0 * sWi[cch]      + x1 * sWi[96 + cch];
      const float iz  = sbi[32 + cch] + x0 * sWi[32 + cch] + x1 * sWi[96 + 32 + cch];
      const float in_ = sbi[64 + cch] + x0 * sWi[64 + cch] + x1 * sWi[96 + 64 + cch];
      const float rg = 1.0f / (1.0f + __expf(-(ir + hr)));
      const float zg = 1.0f / (1.0f + __expf(-(iz + hz)));
      const float ng = tanhf(in_ + rg * hnn);
      hn[cch] = (1.0f - zg) * ng + zg * hc[cch];
    }
    cur = 1 - cur;
  }
  if (active)
    for (int cch = 0; cch < 32; ++cch) Ht[(size_t)i * 32 + cch] = hbuf[cur][tid * 32 + cch];
}

// ---------------------------------------------------------------------------
// Feature assembly: feats[n] = [x[n] (16) | 0 (32, GRU scatter later) | emb (8)]
// ---------------------------------------------------------------------------
__global__ void feats_kernel(const float* __restrict__ x,
                             const int* __restrict__ node_type,
                             const float* __restrict__ type_emb,
                             float* __restrict__ feats, int N) {
  const int gid = blockIdx.x * blockDim.x + threadIdx.x;
  if (gid >= N * 56) return;
  const int n = gid / 56;
  const int c = gid - n * 56;
  float v;
  if (c < 16)      v = x[(size_t)n * 16 + c];
  else if (c < 48) v = 0.0f;
  else             v = type_emb[node_type[n] * 8 + (c - 48)];
  feats[gid] = v;
}

__global__ void scatter_ht_kernel(const float* __restrict__ Ht,
                                  const int* __restrict__ comp_idx,
                                  float* __restrict__ feats, int M) {
  const int gid = blockIdx.x * blockDim.x + threadIdx.x;
  if (gid >= M * 32) return;
  const int i = gid >> 5, c = gid & 31;
  feats[(size_t)comp_idx[i] * 56 + 16 + c] = Ht[gid];
}

// ---------------------------------------------------------------------------
// Small init / edge-attr mean kernels
// ---------------------------------------------------------------------------
__global__ void zero_small_kernel(float* easum, float* gsum) {
  if (threadIdx.x < 8)  easum[threadIdx.x] = 0.0f;
  if (threadIdx.x < 64) gsum[threadIdx.x]  = 0.0f;
}

__global__ void ea_sum_kernel(const float* __restrict__ eattr, float* easum, int E) {
  const int k = threadIdx.x & 7;
  const int slot = (blockIdx.x * blockDim.x + threadIdx.x) >> 3;
  const int nslots = (gridDim.x * blockDim.x) >> 3;
  float s = 0.0f;
  for (int e = slot; e < E; e += nslots) s += eattr[(size_t)e * 8 + k];
  atomicAdd(&easum[k], s);
}

__global__ void ea_final_kernel(const float* easum, float* ea_mean, int E) {
  if (threadIdx.x < 8) ea_mean[threadIdx.x] = easum[threadIdx.x] / (float)E;
}

__global__ void layer_init_kernel(float* amax, float* den, float* agg, int N) {
  const int gid = blockIdx.x * blockDim.x + threadIdx.x;
  if (gid < N * 128) agg[gid] = 0.0f;
  if (gid < N * 2) { amax[gid] = __int_as_float(0xff800000); den[gid] = 0.0f; }
}

// ---------------------------------------------------------------------------
// Edge pass 1: alpha[e][h] = sum_c leaky(xl[s]+xr[d]+ea@We)*att ; segment max.
// 4 threads per edge (32 channels each), pair-reduce per head via shfl_xor.
// ---------------------------------------------------------------------------
__global__ void edge_alpha_kernel(const int* __restrict__ ei,
                                  const float* __restrict__ eattr,
                                  const float* __restrict__ ea_mean,
                                  const float* __restrict__ We,
                                  const float* __restrict__ att,
                                  const float* __restrict__ xl,
                                  const float* __restrict__ xr,
                                  float* __restrict__ alpha,
                                  float* __restrict__ amax,
                                  int E, int Etot) {
  __shared__ float sWe[8 * 128];
  __shared__ float sAtt[128];
  for (int i = threadIdx.x; i < 1024; i += blockDim.x) sWe[i] = We[i];
  for (int i = threadIdx.x; i < 128; i += blockDim.x) sAtt[i] = att[i];
  __syncthreads();

  const int gid = blockIdx.x * blockDim.x + threadIdx.x;
  const int e = gid >> 2;
  if (e >= Etot) return;
  const int q = gid & 3;                 // channel chunk -> head = q>>1
  int s, d; edge_sd(ei, E, e, s, d);

  float ea[8];
#pragma unroll
  for (int k = 0; k < 8; ++k) ea[k] = (e < E) ? eattr[(size_t)e * 8 + k] : ea_mean[k];

  const int c0 = q * 32;
  const float* __restrict__ xls = xl + (size_t)s * 128 + c0;
  const float* __restrict__ xrd = xr + (size_t)d * 128 + c0;
  float part = 0.0f;
  for (int c = 0; c < 32; ++c) {
    const int ch = c0 + c;
    float ee = 0.0f;
#pragma unroll
    for (int k = 0; k < 8; ++k) ee = fmaf(ea[k], sWe[k * 128 + ch], ee);
    float m = xls[c] + xrd[c] + ee;
    m = (m > 0.0f) ? m : 0.2f * m;       // leaky_relu(0.2)
    part = fmaf(m, sAtt[ch], part);
  }
  part += __shfl_xor(part, 1, 32);       // combine 32-ch halves of a head
  if ((q & 1) == 0) {
    const int h = q >> 1;
    alpha[(size_t)e * 2 + h] = part;
    atomicMaxFloat(&amax[d * 2 + h], part);
  }
}

// Edge pass 2: ex = exp(alpha - amax[dst]); segment sum -> den.
__global__ void edge_ex_kernel(const int* __restrict__ ei,
                               float* __restrict__ alpha,
                               const float* __restrict__ amax,
                               float* __restrict__ den,
                               int E, int Etot) {
  const int gid = blockIdx.x * blockDim.x + threadIdx.x;
  if (gid >= Etot * 2) return;
  const int e = gid >> 1, h = gid & 1;
  int s, d; edge_sd(ei, E, e, s, d);
  (void)s;
  const float ex = __expf(alpha[gid] - amax[d * 2 + h]);
  alpha[gid] = ex;
  atomicAdd(&den[d * 2 + h], ex);
}

// Edge pass 3: agg[dst] += xl[src] * (ex / den[dst]).
__global__ void edge_agg_kernel(const int* __restrict__ ei,
                                const float* __restrict__ alpha,
                                const float* __restrict__ den,
                                const float* __restrict__ xl,
                                float* __restrict__ agg,
                                int E, int Etot) {
  const int gid = blockIdx.x * blockDim.x + threadIdx.x;
  const int e = gid >> 2;
  if (e >= Etot) return;
  const int q = gid & 3, h = q >> 1;
  int s, d; edge_sd(ei, E, e, s, d);
  const float w = alpha[(size_t)e * 2 + h] / den[d * 2 + h];
  const int c0 = q * 32;
  const float* __restrict__ xs = xl + (size_t)s * 128 + c0;
  float* __restrict__ ag = agg + (size_t)d * 128 + c0;
  for (int c = 0; c < 32; ++c) atomicAdd(&ag[c], xs[c] * w);
}

// Head mean + bias + optional ELU.  ACT: 0 none, 2 elu.
template <int ACT>
__global__ void gat_final_kernel(const float* __restrict__ agg,
                                 const float* __restrict__ bias,
                                 float* __restrict__ hout, int N) {
  const int gid = blockIdx.x * blockDim.x + threadIdx.x;
  if (gid >= N * 64) return;
  const int n = gid >> 6, c = gid & 63;
  float v = 0.5f * (agg[(size_t)n * 128 + c] + agg[(size_t)n * 128 + 64 + c]) + bias[c];
  if (ACT == 2) v = (v > 0.0f) ? v : (__expf(v) - 1.0f);
  hout[gid] = v;
}

// Column sum of h [N,64] into gsum[64] (block-local LDS reduce first).
__global__ void colsum_kernel(const float* __restrict__ h, float* gsum, int N) {
  __shared__ float s[64];
  if (threadIdx.x < 64) s[threadIdx.x] = 0.0f;
  __syncthreads();
  const int c = threadIdx.x & 63;
  const int rowBase = (blockIdx.x * blockDim.x + threadIdx.x) >> 6;
  const int stride = (gridDim.x * blockDim.x) >> 6;
  float loc = 0.0f;
  for (int n = rowBase; n < N; n += stride) loc += h[(size_t)n * 64 + c];
  atomicAdd(&s[c], loc);
  __syncthreads();
  if (threadIdx.x < 64) atomicAdd(&gsum[threadIdx.x], s[threadIdx.x]);
}

// g = tanh(mean(h) @ Wg + bg); mu_t head; ls_t.  One block of 64 threads.
__global__ void head_kernel(const float* __restrict__ gsum,
                            const float* __restrict__ Wg, const float* __restrict__ bg,
                            const float* __restrict__ Wt0, const float* __restrict__ bt0,
                            const float* __restrict__ Wt1, const float* __restrict__ bt1,
                            const float* __restrict__ logstd_theta,
                            float* __restrict__ gvec, float* __restrict__ out,
                            int N, int Mc) {
  __shared__ float gmean[64];
  __shared__ float g[64];
  __shared__ float h2t[64];
  const int c = threadIdx.x;
  gmean[c] = gsum[c] / (float)N;
  __syncthreads();
  float acc = bg[c];
  for (int k = 0; k < 64; ++k) acc = fmaf(gmean[k], Wg[k * 64 + c], acc);
  g[c] = tanhf(acc);
  __syncthreads();
  gvec[c] = g[c];
  float acc2 = bt0[c];
  for (int k = 0; k < 64; ++k) acc2 = fmaf(g[k], Wt0[k * 64 + c], acc2);
  h2t[c] = fmaxf(acc2, 0.0f);
  __syncthreads();
  if (c < 6) {
    float acc3 = bt1[c];
    for (int k = 0; k < 64; ++k) acc3 = fmaf(h2t[k], Wt1[k * 6 + c], acc3);
    out[2 * Mc + c]     = acc3;              // mu_t
    out[2 * Mc + 6 + c] = logstd_theta[c];   // ls_t
  }
}

// z[i] = [ h[comp_idx[i]] (64) | g (64) ]
__global__ void buildz_kernel(const float* __restrict__ h,
                              const int* __restrict__ comp_idx,
                              const float* __restrict__ gvec,
                              float* __restrict__ z, int M) {
  const int gid = blockIdx.x * blockDim.x + threadIdx.x;
  if (gid >= M * 128) return;
  const int i = gid >> 7, c = gid & 127;
  z[gid] = (c < 64) ? h[(size_t)comp_idx[i] * 64 + c] : gvec[c - 64];
}

// mu_l = h1c @ W1[64,1] + b1 ; ls_l broadcast.
__global__ void mul_kernel(const float* __restrict__ h1c,
                           const float* __restrict__ W1,
                           const float* __restrict__ b1,
                           const float* __restrict__ logstd_lambda,
                           float* __restrict__ out, int M) {
  __shared__ float sW[64];
  __shared__ float sb[2];
  if (threadIdx.x < 64) sW[threadIdx.x] = W1[threadIdx.x];
  if (threadIdx.x == 0) { sb[0] = b1[0]; sb[1] = logstd_lambda[0]; }
  __syncthreads();
  const int i = blockIdx.x * blockDim.x + threadIdx.x;
  if (i >= M) return;
  float acc = sb[0];
  for (int k = 0; k < 64; ++k) acc = fmaf(h1c[(size_t)i * 64 + k], sW[k], acc);
  out[i] = acc;       // mu_l
  out[M + i] = sb[1]; // ls_l
}

// ---------------------------------------------------------------------------
static inline unsigned cdiv(unsigned a, unsigned b) { return (a + b - 1) / b; }

extern "C" void kernel_launch(void* const* d_in, const int* in_sizes, int n_in,
                              void* d_out, int out_size, void* d_ws, size_t ws_size,
                              hipStream_t stream) {
  // ---- inputs (recursive insertion-order flatten of setup_inputs dict) ----
  const float* x         = (const float*)d_in[0];
  const int*   node_type = (const int*)  d_in[1];
  const int*   ei        = (const int*)  d_in[2];   // [2,E]
  const float* eattr     = (const float*)d_in[3];
  const float* seq       = (const float*)d_in[4];
  const int*   comp_idx  = (const int*)  d_in[5];
  const float* gWi = (const float*)d_in[6];
  const float* gWh = (const float*)d_in[7];
  const float* gbi = (const float*)d_in[8];
  const float* gbh = (const float*)d_in[9];
  const float* type_emb = (const float*)d_in[10];
  const float* linW = (const float*)d_in[11];
  const float* linb = (const float*)d_in[12];
  const float* g1Wl = (const float*)d_in[13];
  const float* g1bl = (const float*)d_in[14];
  const float* g1Wr = (const float*)d_in[15];
  const float* g1br = (const float*)d_in[16];
  const float* g1We = (const float*)d_in[17];
  const float* g1att = (const float*)d_in[18];
  const float* g1bias = (const float*)d_in[19];
  const float* g2Wl = (const float*)d_in[20];
  const float* g2bl = (const float*)d_in[21];
  const float* g2Wr = (const float*)d_in[22];
  const float* g2br = (const float*)d_in[23];
  const float* g2We = (const float*)d_in[24];
  const float* g2att = (const float*)d_in[25];
  const float* g2bias = (const float*)d_in[26];
  const float* lgW = (const float*)d_in[27];
  const float* lgb = (const float*)d_in[28];
  const float* ml0W = (const float*)d_in[29];
  const float* ml0b = (const float*)d_in[30];
  const float* ml1W = (const float*)d_in[31];
  const float* ml1b = (const float*)d_in[32];
  const float* mt0W = (const float*)d_in[33];
  const float* mt0b = (const float*)d_in[34];
  const float* mt1W = (const float*)d_in[35];
  const float* mt1b = (const float*)d_in[36];
  const float* logstd_lambda = (const float*)d_in[37];
  const float* logstd_theta  = (const float*)d_in[38];

  const int N = in_sizes[0] / 16;     // 50000
  const int E = in_sizes[2] / 2;      // 800000
  const int Mc = in_sizes[5];         // 4096
  const int Etot = E + N;             // with self loops
  float* out = (float*)d_out;

  // ---- workspace carve-up (floats) ----
  float* ws = (float*)d_ws;
  size_t o = 0;
  float* easum = ws + o; o += 8;
  float* eamean = ws + o; o += 8;
  float* gsum = ws + o; o += 64;
  float* gvec = ws + o; o += 64;
  float* Ht = ws + o; o += (size_t)Mc * 32;
  float* feats = ws + o; o += (size_t)N * 56;
  float* hA = ws + o; o += (size_t)N * 64;
  float* hB = ws + o; o += (size_t)N * 64;
  float* xl = ws + o; o += (size_t)N * 128;
  float* xr = ws + o; o += (size_t)N * 128;
  float* alpha = ws + o; o += (size_t)Etot * 2;
  float* amax = ws + o; o += (size_t)N * 2;
  float* den = ws + o; o += (size_t)N * 2;
  float* agg = ws + o; o += (size_t)N * 128;
  float* zbuf = ws + o; o += (size_t)Mc * 128;
  float* h1c = ws + o; o += (size_t)Mc * 64;
  (void)ws_size; (void)n_in; (void)out_size;

  const int TB = 256;
  // ---- stage 0: small init, edge-attr mean, GRU, features ----
  zero_small_kernel<<<1, 64, 0, stream>>>(easum, gsum);
  ea_sum_kernel<<<512, TB, 0, stream>>>(eattr, easum, E);
  ea_final_kernel<<<1, 8, 0, stream>>>(easum, eamean, E);
  gru_kernel<<<cdiv(Mc, 128), 128, 0, stream>>>(seq, gWi, gWh, gbi, gbh, Ht, Mc);
  feats_kernel<<<cdiv(N * 56, TB), TB, 0, stream>>>(x, node_type, type_emb, feats, N);
  scatter_ht_kernel<<<cdiv(Mc * 32, TB), TB, 0, stream>>>(Ht, comp_idx, feats, Mc);

  // ---- lin_in: h0 = relu(feats @ W + b)  [N,56]@[56,64] ----
  wmma_gemm_kernel<1, 56, 64><<<dim3(cdiv(N, 256), 4), 128, 0, stream>>>(feats, linW, linb, hA, N);

  // ---- GAT layer 1 (input hA, output hB with ELU) ----
  wmma_gemm_kernel<0, 64, 128><<<dim3(cdiv(N, 256), 8), 128, 0, stream>>>(hA, g1Wl, g1bl, xl, N);
  wmma_gemm_kernel<0, 64, 128><<<dim3(cdiv(N, 256), 8), 128, 0, stream>>>(hA, g1Wr, g1br, xr, N);
  layer_init_kernel<<<cdiv(N * 128, TB), TB, 0, stream>>>(amax, den, agg, N);
  edge_alpha_kernel<<<cdiv(Etot * 4, TB), TB, 0, stream>>>(ei, eattr, eamean, g1We, g1att,
                                                           xl, xr, alpha, amax, E, Etot);
  edge_ex_kernel<<<cdiv(Etot * 2, TB), TB, 0, stream>>>(ei, alpha, amax, den, E, Etot);
  edge_agg_kernel<<<cdiv(Etot * 4, TB), TB, 0, stream>>>(ei, alpha, den, xl, agg, E, Etot);
  gat_final_kernel<2><<<cdiv(N * 64, TB), TB, 0, stream>>>(agg, g1bias, hB, N);

  // ---- GAT layer 2 (input hB, output hA, no activation) ----
  wmma_gemm_kernel<0, 64, 128><<<dim3(cdiv(N, 256), 8), 128, 0, stream>>>(hB, g2Wl, g2bl, xl, N);
  wmma_gemm_kernel<0, 64, 128><<<dim3(cdiv(N, 256), 8), 128, 0, stream>>>(hB, g2Wr, g2br, xr, N);
  layer_init_kernel<<<cdiv(N * 128, TB), TB, 0, stream>>>(amax, den, agg, N);
  edge_alpha_kernel<<<cdiv(Etot * 4, TB), TB, 0, stream>>>(ei, eattr, eamean, g2We, g2att,
                                                           xl, xr, alpha, amax, E, Etot);
  edge_ex_kernel<<<cdiv(Etot * 2, TB), TB, 0, stream>>>(ei, alpha, amax, den, E, Etot);
  edge_agg_kernel<<<cdiv(Etot * 4, TB), TB, 0, stream>>>(ei, alpha, den, xl, agg, E, Etot);
  gat_final_kernel<0><<<cdiv(N * 64, TB), TB, 0, stream>>>(agg, g2bias, hA, N);

  // ---- graph readout + heads ----
  colsum_kernel<<<256, TB, 0, stream>>>(hA, gsum, N);
  head_kernel<<<1, 64, 0, stream>>>(gsum, lgW, lgb, mt0W, mt0b, mt1W, mt1b,
                                    logstd_theta, gvec, out, N, Mc);
  buildz_kernel<<<cdiv(Mc * 128, TB), TB, 0, stream>>>(hA, comp_idx, gvec, zbuf, Mc);
  wmma_gemm_kernel<1, 128, 64><<<dim3(cdiv(Mc, 256), 4), 128, 0, stream>>>(zbuf, ml0W, ml0b, h1c, Mc);
  mul_kernel<<<cdiv(Mc, TB), TB, 0, stream>>>(h1c, ml1W, ml1b, logstd_lambda, out, Mc);
}